// BatchChannelDecorrelationLoss_79388175499363
// MI455X (gfx1250) — compile-verified
//
#include <hip/hip_runtime.h>
#include <math.h>
#include <stdint.h>

// ---------------- problem constants (from reference) ----------------
#define N_BATCH   16
#define C_ALL     192
#define HY        32
#define WY        32
#define CHW_Y     (HY * WY)               // 1024
#define SAMP      (N_BATCH * CHW_Y)       // 16384 samples per channel
#define TOPK      64
#define IMG_ELEMS (16 * 3 * 512 * 512)    // 12582912
#define LAT_ELEMS (N_BATCH * C_ALL * CHW_Y) // 3145728
#define NUM_PIXELS (16 * 512 * 512)       // 4194304

#define MSE_BLOCKS 1024
#define BPP_BLOCKS 256
#define KCHUNK     128                    // samples staged in LDS per chunk

typedef float    v2f  __attribute__((ext_vector_type(2)));
typedef float    v8f  __attribute__((ext_vector_type(8)));

__device__ __forceinline__ void wait_asynccnt0() {
#if __has_builtin(__builtin_amdgcn_s_wait_asynccnt)
    __builtin_amdgcn_s_wait_asynccnt(0);
#else
    asm volatile("s_wait_asynccnt 0x0" ::: "memory");
#endif
}

// ---------------- kernel 1: MSE partial sums (float4 / b128 loads) ----------------
__global__ void k_mse(const float4* __restrict__ xh, const float4* __restrict__ tg,
                      float* __restrict__ out) {
    __shared__ float s[256];
    const int n4 = IMG_ELEMS / 4;
    float acc = 0.f;
    for (int i = blockIdx.x * blockDim.x + threadIdx.x; i < n4;
         i += gridDim.x * blockDim.x) {
        const float4 a = xh[i];
        const float4 b = tg[i];
        float d0 = a.x - b.x, d1 = a.y - b.y, d2 = a.z - b.z, d3 = a.w - b.w;
        acc += d0 * d0 + d1 * d1 + d2 * d2 + d3 * d3;
    }
    s[threadIdx.x] = acc;
    __syncthreads();
    for (int o = 128; o > 0; o >>= 1) {
        if (threadIdx.x < o) s[threadIdx.x] += s[threadIdx.x + o];
        __syncthreads();
    }
    if (threadIdx.x == 0) out[blockIdx.x] = s[0];
}

// ---------------- kernel 2: sum(log(likelihood)) partials (float4) ----------------
__global__ void k_bpp(const float4* __restrict__ lik, float* __restrict__ out) {
    __shared__ float s[256];
    const int n4 = LAT_ELEMS / 4;
    float acc = 0.f;
    for (int i = blockIdx.x * blockDim.x + threadIdx.x; i < n4;
         i += gridDim.x * blockDim.x) {
        const float4 a = lik[i];
        acc += __logf(a.x) + __logf(a.y) + __logf(a.z) + __logf(a.w);
    }
    s[threadIdx.x] = acc;
    __syncthreads();
    for (int o = 128; o > 0; o >>= 1) {
        if (threadIdx.x < o) s[threadIdx.x] += s[threadIdx.x + o];
        __syncthreads();
    }
    if (threadIdx.x == 0) out[blockIdx.x] = s[0];
}

// ---------------- kernel 3: per-(n,c) range of rounded latent ----------------
__global__ void k_range(const float* __restrict__ y, int* __restrict__ out) {
    const int nc = blockIdx.x;                 // n*C_ALL + c
    const float* p = y + (size_t)nc * CHW_Y;
    int mn = 0x7fffffff, mx = (int)0x80000000;
    for (int i = threadIdx.x; i < CHW_Y; i += blockDim.x) {
        int v = (int)rintf(p[i]);              // round-half-even, matches jnp.round
        mn = v < mn ? v : mn;
        mx = v > mx ? v : mx;
    }
    __shared__ int smn[256], smx[256];
    smn[threadIdx.x] = mn;
    smx[threadIdx.x] = mx;
    __syncthreads();
    for (int o = 128; o > 0; o >>= 1) {
        if (threadIdx.x < o) {
            int a = smn[threadIdx.x + o];
            int b = smx[threadIdx.x + o];
            if (a < smn[threadIdx.x]) smn[threadIdx.x] = a;
            if (b > smx[threadIdx.x]) smx[threadIdx.x] = b;
        }
        __syncthreads();
    }
    if (threadIdx.x == 0) out[nc] = smx[0] - smn[0];
}

// ---------------- kernel 4: rates + top-64 by rank ----------------
// argsort(rates)[::-1][:64]  (stable ascending, reversed => ties: larger index first)
__global__ void k_topk(const int* __restrict__ range, int* __restrict__ rates,
                       int* __restrict__ idx) {
    __shared__ int sr[C_ALL];
    const int c = threadIdx.x;
    if (c < C_ALL) {
        int s = 0;
        for (int n = 0; n < N_BATCH; ++n) s += range[n * C_ALL + c];
        sr[c] = s;
        rates[c] = s;
    }
    __syncthreads();
    if (c < C_ALL) {
        const int mine = sr[c];
        int rank = 0;
        for (int j = 0; j < C_ALL; ++j) {
            const int rj = sr[j];
            rank += (rj > mine) || (rj == mine && j > c);
        }
        if (rank < TOPK) idx[rank] = c;
    }
}

// ---------------- kernel 5: per-selected-channel mean ----------------
__global__ void k_mean(const float* __restrict__ y, const int* __restrict__ idx,
                       float* __restrict__ mean) {
    const int ch = idx[blockIdx.x];
    float acc = 0.f;
    for (int m = threadIdx.x; m < SAMP; m += blockDim.x) {
        acc += y[(size_t)(m >> 10) * (C_ALL * CHW_Y) + (size_t)ch * CHW_Y + (m & 1023)];
    }
    __shared__ float s[256];
    s[threadIdx.x] = acc;
    __syncthreads();
    for (int o = 128; o > 0; o >>= 1) {
        if (threadIdx.x < o) s[threadIdx.x] += s[threadIdx.x + o];
        __syncthreads();
    }
    if (threadIdx.x == 0) mean[blockIdx.x] = s[0] * (1.0f / (float)SAMP);
}

// ---------------- kernel 6: 64x64 Gram via fp32 WMMA + async double-buffer ----
// One workgroup, 16 waves; wave w owns output tile (w/4, w%4).
// Uncentered Gram G = Y^T Y accumulated by v_wmma_f32_16x16x4_f32;
// cov = (G - M*mu_i*mu_j)/(M-1) applied in the epilogue.
// Global->LDS staging uses GLOBAL_LOAD_ASYNC_TO_LDS_B32 (ASYNCcnt) with
// double-buffered LDS so the next chunk's copy overlaps this chunk's WMMAs.
__global__ void __launch_bounds__(512) k_cov(const float* __restrict__ y,
                                             const int* __restrict__ idx,
                                             const float* __restrict__ mean,
                                             float* __restrict__ cov) {
    __shared__ float sY[2][KCHUNK * 64];   // 2 x 32 KB: [k][c] raw values
    __shared__ int   sIdx[64];
    __shared__ float sMean[64];

    const int tid = threadIdx.x;
    if (tid < 64) {
        sIdx[tid]  = idx[tid];
        sMean[tid] = mean[tid];
    }
    __syncthreads();

    const int wave = tid >> 5;
    const int lane = tid & 31;
    const int i0 = (wave >> 2) * 16;           // tile row base (channels)
    const int j0 = (wave & 3) * 16;            // tile col base (channels)
    const int half = lane >> 4;                // 0: lanes 0-15, 1: lanes 16-31
    const int mA = i0 + (lane & 15);           // A-matrix row  = channel i
    const int nB = j0 + (lane & 15);           // B-matrix col  = channel j
    const int kb = half * 2;                   // f32 16x4 A/B: lanes 0-15 K={0,1}, 16-31 K={2,3}

    // staging geometry: each thread owns a fixed channel c, samples k = kBase + 8*j
    const int cMy   = tid & 63;
    const int kBase = tid >> 6;
    const size_t chanOff = (size_t)sIdx[cMy] * CHW_Y;

    v8f acc = {};

#define ISSUE_CHUNK(S0, BUF)                                                        \
    do {                                                                            \
        _Pragma("unroll")                                                           \
        for (int j = 0; j < (KCHUNK * 64) / 512; ++j) {                             \
            const int k = kBase + 8 * j;                                            \
            const int m = (S0) + k;                                                 \
            const float* gptr = y + (size_t)(m >> 10) * (C_ALL * CHW_Y)             \
                                  + chanOff + (m & 1023);                           \
            const uint32_t ldsa = (uint32_t)(uintptr_t)(&sY[(BUF)][k * 64 + cMy]);  \
            asm volatile("global_load_async_to_lds_b32 %0, %1, off"                 \
                         :: "v"(ldsa), "v"((uint64_t)(uintptr_t)gptr)               \
                         : "memory");                                               \
        }                                                                           \
    } while (0)

    // prologue: fill buffer 0
    ISSUE_CHUNK(0, 0);
    wait_asynccnt0();
    __syncthreads();

    int buf = 0;
    for (int s0 = 0; s0 < SAMP; s0 += KCHUNK) {
        if (s0 + KCHUNK < SAMP) ISSUE_CHUNK(s0 + KCHUNK, buf ^ 1);   // overlap copy

        const float* sb = sY[buf];
#pragma unroll 8
        for (int k4 = 0; k4 < KCHUNK; k4 += 4) {
            v2f a, b;
            a.x = sb[(k4 + kb) * 64 + mA];
            a.y = sb[(k4 + kb + 1) * 64 + mA];
            b.x = sb[(k4 + kb) * 64 + nB];
            b.y = sb[(k4 + kb + 1) * 64 + nB];
            acc = __builtin_amdgcn_wmma_f32_16x16x4_f32(
                false, a, false, b, (short)0, acc, false, false);
        }

        wait_asynccnt0();     // my async writes for next buffer have landed
        __syncthreads();      // everyone's writes landed; everyone done reading buf
        buf ^= 1;
    }
#undef ISSUE_CHUNK

    // C/D layout: VGPR r -> M = r + half*8; N = lane%16
    const float inv = 1.0f / (float)(SAMP - 1);
    const int Mb = half * 8;
    const int Nc = lane & 15;
    const float muJ = sMean[j0 + Nc];
#pragma unroll
    for (int r = 0; r < 8; ++r) {
        const float muI = sMean[i0 + Mb + r];
        cov[(i0 + Mb + r) * 64 + (j0 + Nc)] =
            (acc[r] - (float)SAMP * muI * muJ) * inv;
    }
}

// ---------------- kernel 7: combine scalar loss ----------------
__global__ void k_final(const float* __restrict__ pm, const float* __restrict__ pb,
                        const float* __restrict__ cov, float* __restrict__ out) {
    __shared__ float s[256];
    const int tid = threadIdx.x;

    float a = 0.f;
    for (int i = tid; i < MSE_BLOCKS; i += 256) a += pm[i];
    s[tid] = a;
    __syncthreads();
    for (int o = 128; o > 0; o >>= 1) { if (tid < o) s[tid] += s[tid + o]; __syncthreads(); }
    const float mse_sum = s[0];
    __syncthreads();

    float b = 0.f;
    for (int i = tid; i < BPP_BLOCKS; i += 256) b += pb[i];
    s[tid] = b;
    __syncthreads();
    for (int o = 128; o > 0; o >>= 1) { if (tid < o) s[tid] += s[tid + o]; __syncthreads(); }
    const float log_sum = s[0];
    __syncthreads();

    float c = 0.f;
    for (int i = tid; i < TOPK * TOPK; i += 256) {
        const int r = i >> 6, cc = i & 63;
        if (r != cc) { const float v = cov[i]; c += v * v; }
    }
    s[tid] = c;
    __syncthreads();
    for (int o = 128; o > 0; o >>= 1) { if (tid < o) s[tid] += s[tid + o]; __syncthreads(); }

    if (tid == 0) {
        const float mse  = mse_sum * (1.0f / (float)IMG_ELEMS);
        const float bpp  = log_sum / (-0.69314718055994531f * (float)NUM_PIXELS);
        const float corr = s[0];
        out[0] = 0.01f * 65025.0f * mse + bpp + 1e-4f * corr;
    }
}

// ---------------- launch ----------------
extern "C" void kernel_launch(void* const* d_in, const int* in_sizes, int n_in,
                              void* d_out, int out_size, void* d_ws, size_t ws_size,
                              hipStream_t stream) {
    const float* y      = (const float*)d_in[0];
    const float* x_hat  = (const float*)d_in[1];
    const float* target = (const float*)d_in[2];
    const float* lik    = (const float*)d_in[3];
    float* out = (float*)d_out;

    // workspace carve-up (~35 KB)
    float* ws_mse   = (float*)d_ws;                 // [1024]
    float* ws_bpp   = ws_mse + MSE_BLOCKS;          // [256]
    int*   ws_range = (int*)(ws_bpp + BPP_BLOCKS);  // [3072]
    int*   ws_rates = ws_range + N_BATCH * C_ALL;   // [192]
    int*   ws_idx   = ws_rates + C_ALL;             // [64]
    float* ws_mean  = (float*)(ws_idx + TOPK);      // [64]
    float* ws_cov   = ws_mean + TOPK;               // [4096]

    k_mse  <<<MSE_BLOCKS, 256, 0, stream>>>((const float4*)x_hat, (const float4*)target, ws_mse);
    k_bpp  <<<BPP_BLOCKS, 256, 0, stream>>>((const float4*)lik, ws_bpp);
    k_range<<<N_BATCH * C_ALL, 256, 0, stream>>>(y, ws_range);
    k_topk <<<1, C_ALL, 0, stream>>>(ws_range, ws_rates, ws_idx);
    k_mean <<<TOPK, 256, 0, stream>>>(y, ws_idx, ws_mean);
    k_cov  <<<1, 512, 0, stream>>>(y, ws_idx, ws_mean, ws_cov);
    k_final<<<1, 256, 0, stream>>>(ws_mse, ws_bpp, ws_cov, out);
}